// NodeWeightAndSum_68745246539864
// MI455X (gfx1250) — compile-verified
//
#include <hip/hip_runtime.h>
#include <hip/hip_bf16.h>

typedef __attribute__((ext_vector_type(2))) float v2f;
typedef __attribute__((ext_vector_type(4))) float v4f;
typedef __attribute__((ext_vector_type(8))) float v8f;

#define FEAT 128
#define WAVES_TOTAL 8192
#define NBLOCKS 1024
#define TPB 256

__device__ __forceinline__ void flush_acc(float* __restrict__ hsum,
                                          long long segid, int lane,
                                          const v4f& acc) {
    float* dst = hsum + (size_t)segid * FEAT + lane * 4;
    unsafeAtomicAdd(dst + 0, acc.x);
    unsafeAtomicAdd(dst + 1, acc.y);
    unsafeAtomicAdd(dst + 2, acc.z);
    unsafeAtomicAdd(dst + 3, acc.w);
}

__global__ __launch_bounds__(TPB)
void NodeWeightAndSum_kernel(const float* __restrict__ x,         // [N,128]
                             const long long* __restrict__ seg,   // [N]
                             const float* __restrict__ Wg,        // [128]
                             const float* __restrict__ bg,        // [1]
                             float* __restrict__ hsum,            // [G,128] (pre-zeroed)
                             float* __restrict__ wout,            // [N]
                             int N) {
    const int lane = threadIdx.x & 31;
    const int wave = blockIdx.x * (TPB >> 5) + (threadIdx.x >> 5);

    int chunk = ((N + WAVES_TOTAL - 1) / WAVES_TOTAL + 15) & ~15;
    long nBeg = (long)wave * chunk;
    if (nBeg >= N) return;
    long nEnd = nBeg + chunk; if (nEnd > N) nEnd = N;

    const float bias = bg[0];
    const int koff = (lane >> 4) * 2;   // K offset within a 4-chunk: 0 or 2
    const int row  = lane & 15;         // node row (A-matrix M) this lane feeds

    // Preload W in WMMA B-matrix layout (4x16, all 16 columns identical):
    // per kk: B VGPR0/1 = W[4kk+koff], W[4kk+koff+1]
    v2f Wv[32];
    #pragma unroll
    for (int kk = 0; kk < 32; ++kk)
        Wv[kk] = *(const v2f*)(Wg + kk * 4 + koff);

    // Per-lane W slice for the scalar tail path
    const v4f W4 = *(const v4f*)(Wg + lane * 4);

    v4f acc = {0.f, 0.f, 0.f, 0.f};
    long long curSeg = -1;

    long i = nBeg;
    for (; i + 16 <= nEnd; i += 16) {
        const float* base = x + (size_t)i * FEAT;
        const float* arow = base + (size_t)row * FEAT + koff;

        // ---- phase 1: batch all 32 A-tile loads (32 loads in flight) ----
        v2f a[32];
        #pragma unroll
        for (int kk = 0; kk < 32; ++kk)
            a[kk] = *(const v2f*)(arow + kk * 4);

        // cooperative segment-id load: lane l holds seg[i + (l&15)]
        long long sidv = seg[i + row];

        // prefetch next 16-node block (global_prefetch_b8)
        __builtin_prefetch(base + 16 * FEAT + lane * 32, 0, 0);

        // ---- phase 2: 32 accumulating V_WMMA_F32_16X16X4_F32 ----
        v8f c = {0.f, 0.f, 0.f, 0.f, 0.f, 0.f, 0.f, 0.f};
        #pragma unroll
        for (int kk = 0; kk < 32; ++kk)
            c = __builtin_amdgcn_wmma_f32_16x16x4_f32(
                    false, a[kk], false, Wv[kk], (short)0, c, false, false);

        // C layout: lanes 0-15 VGPR r = dot(node r); lanes 16-31 VGPR r = dot(node r+8)
        float sg[8];
        #pragma unroll
        for (int r = 0; r < 8; ++r)
            sg[r] = 1.0f / (1.0f + __expf(-(c[r] + bias)));

        // store weights: lane 0 -> nodes i..i+7, lane 16 -> nodes i+8..i+15
        if (row == 0) {
            long off = i + (long)((lane >> 4) * 8);
            v4f w0 = {sg[0], sg[1], sg[2], sg[3]};
            v4f w1 = {sg[4], sg[5], sg[6], sg[7]};
            *(v4f*)(wout + off)     = w0;
            *(v4f*)(wout + off + 4) = w1;
        }

        // ---- phase 3: weighted segmented accumulation ----
        long long s0 = __shfl(sidv, 0);
        if (__all(sidv == s0)) {
            // fast path: whole block in one segment (avg run ~122 nodes)
            if (s0 != curSeg) {
                if (curSeg >= 0) flush_acc(hsum, curSeg, lane, acc);
                curSeg = s0;
                acc.x = 0.f; acc.y = 0.f; acc.z = 0.f; acc.w = 0.f;
            }
            #pragma unroll
            for (int m = 0; m < 16; ++m) {
                float wm = __shfl(sg[m & 7], (m < 8) ? 0 : 16);
                v4f xm = *(const v4f*)(base + (size_t)m * FEAT + lane * 4);
                acc.x += wm * xm.x; acc.y += wm * xm.y;
                acc.z += wm * xm.z; acc.w += wm * xm.w;
            }
        } else {
            // slow path: segment boundary inside the block
            #pragma unroll
            for (int m = 0; m < 16; ++m) {
                float wm = __shfl(sg[m & 7], (m < 8) ? 0 : 16);
                long long sid = __shfl(sidv, m);
                v4f xm = *(const v4f*)(base + (size_t)m * FEAT + lane * 4);
                if (sid != curSeg) {
                    if (curSeg >= 0) flush_acc(hsum, curSeg, lane, acc);
                    curSeg = sid;
                    acc.x = wm * xm.x; acc.y = wm * xm.y;
                    acc.z = wm * xm.z; acc.w = wm * xm.w;
                } else {
                    acc.x += wm * xm.x; acc.y += wm * xm.y;
                    acc.z += wm * xm.z; acc.w += wm * xm.w;
                }
            }
        }
    }

    // ---- tail: one node per step, wave-wide shuffle reduction ----
    for (; i < nEnd; ++i) {
        v4f xm = *(const v4f*)(x + (size_t)i * FEAT + lane * 4);
        float p = xm.x * W4.x + xm.y * W4.y + xm.z * W4.z + xm.w * W4.w;
        #pragma unroll
        for (int s = 16; s >= 1; s >>= 1) p += __shfl_xor(p, s);
        float wm = 1.0f / (1.0f + __expf(-(p + bias)));
        if (lane == 0) wout[i] = wm;
        long long sid = seg[i];
        if (sid != curSeg) {
            if (curSeg >= 0) flush_acc(hsum, curSeg, lane, acc);
            curSeg = sid;
            acc.x = wm * xm.x; acc.y = wm * xm.y;
            acc.z = wm * xm.z; acc.w = wm * xm.w;
        } else {
            acc.x += wm * xm.x; acc.y += wm * xm.y;
            acc.z += wm * xm.z; acc.w += wm * xm.w;
        }
    }

    if (curSeg >= 0) flush_acc(hsum, curSeg, lane, acc);
}

extern "C" void kernel_launch(void* const* d_in, const int* in_sizes, int n_in,
                              void* d_out, int out_size, void* d_ws, size_t ws_size,
                              hipStream_t stream) {
    const float*     x   = (const float*)d_in[0];
    const long long* seg = (const long long*)d_in[1];
    // d_in[2] = num_graphs (scalar, unused on device)
    const float*     Wg  = (const float*)d_in[3];
    const float*     bg  = (const float*)d_in[4];

    const int N = in_sizes[0] / FEAT;
    float* out  = (float*)d_out;                    // [G*128] then [N]
    const size_t hElems = (size_t)out_size - (size_t)N;
    float* wout = out + hElems;

    // zero the segment-sum region (accumulated via f32 atomics); weights fully overwritten
    hipMemsetAsync(out, 0, hElems * sizeof(float), stream);

    NodeWeightAndSum_kernel<<<NBLOCKS, TPB, 0, stream>>>(x, seg, Wg, bg, out, wout, N);
}